// RewireNetGraphClassification_14259291423164
// MI455X (gfx1250) — compile-verified
//
#include <hip/hip_runtime.h>
#include <math.h>

typedef float v2f __attribute__((ext_vector_type(2)));
typedef float v8f __attribute__((ext_vector_type(8)));

#define B_GRAPHS 64
#define N_NODES  128
#define F0_DIM   32
#define R_DIM    1118
#define D_DIM    1150   // F0 + R
#define E_EDGES  512
#define NFILT    21
#define NN       (N_NODES * N_NODES)
#define A_CONST  6.25e-6f          // (STEP/2)^4 = 0.05^4
#define SQRT2F   1.41421356237309515f
#define EPSV     1e-12f

// ---------------------------------------------------------------------------
// fp32 WMMA 16x16x4 wrapper (CDNA5: V_WMMA_F32_16X16X4_F32)
// 8-arg pattern: (neg_a, A, neg_b, B, c_mod, C, reuse_a, reuse_b)
// ---------------------------------------------------------------------------
static __device__ __forceinline__ v8f wmma4(v2f a, v2f b, v8f c) {
  return __builtin_amdgcn_wmma_f32_16x16x4_f32(false, a, false, b, (short)0, c,
                                               false, false);
}

// 128x128x128 fp32 GEMM, A/B in LDS, D written to Dst with row stride
// dstStride; diagAdd added on the diagonal. 256 threads = 8 waves, each wave
// owns 8 of the 64 16x16 output tiles.
static __device__ __forceinline__ void gemm128_lds(const float* __restrict__ A,
                                                   const float* __restrict__ Bm,
                                                   float* __restrict__ Dst,
                                                   int dstStride, float diagAdd,
                                                   int tid) {
  const int lane = tid & 31, wave = tid >> 5;
  const int l16 = lane & 15, hi = lane >> 4;
  for (int t8 = 0; t8 < 8; ++t8) {
    const int t = wave * 8 + t8;
    const int mt = (t >> 3) * 16, nt = (t & 7) * 16;
    v8f acc = {};
    for (int kk = 0; kk < 32; ++kk) {
      const int kb = kk * 4 + 2 * hi;
      v2f a, b;
      const int r = mt + l16;
      a.x = A[r * 128 + kb];
      a.y = A[r * 128 + kb + 1];
      const int cc = nt + l16;
      b.x = Bm[kb * 128 + cc];
      b.y = Bm[(kb + 1) * 128 + cc];
      acc = wmma4(a, b, acc);
    }
    for (int v = 0; v < 8; ++v) {
      const int row = mt + v + 8 * hi;
      const int col = nt + l16;
      float val = acc[v];
      if (row == col) val += diagAdd;
      Dst[row * dstStride + col] = val;
    }
  }
}

// ---------------------------------------------------------------------------
// 1) feats = concat(x, rs * sqrt(1/R))
// ---------------------------------------------------------------------------
__global__ void k_feats(const float* __restrict__ x, const float* __restrict__ rs,
                        float* __restrict__ feats) {
  const float scale = sqrtf(1.0f / (float)R_DIM);
  const size_t total = (size_t)B_GRAPHS * N_NODES * D_DIM;
  for (size_t idx = blockIdx.x * 256ull + threadIdx.x; idx < total;
       idx += (size_t)gridDim.x * 256ull) {
    const size_t bn = idx / D_DIM;
    const int d = (int)(idx % D_DIM);
    feats[idx] = (d < F0_DIM) ? x[bn * F0_DIM + d]
                              : rs[bn * R_DIM + (d - F0_DIM)] * scale;
  }
}

// ---------------------------------------------------------------------------
// 2) symmetric-normalized Laplacian per graph (1 block / graph, 128 thr)
// ---------------------------------------------------------------------------
__global__ void k_lap(const int* __restrict__ esrc, const int* __restrict__ edst,
                      float* __restrict__ L) {
  extern __shared__ float smem[];
  float* A = smem;          // 16384
  float* dinv = smem + NN;  // 128
  const int g = blockIdx.x, tid = threadIdx.x;
  for (int i = tid; i < NN; i += 128) A[i] = 0.0f;
  __syncthreads();
  for (int e = tid; e < E_EDGES; e += 128) {
    const int s = esrc[g * E_EDGES + e] & (N_NODES - 1);
    const int d = edst[g * E_EDGES + e] & (N_NODES - 1);
    A[s * N_NODES + d] = 1.0f;   // A = max(A, A^T) for 0/1 weights
    A[d * N_NODES + s] = 1.0f;
  }
  __syncthreads();
  float deg = 0.0f;
  for (int j = 0; j < N_NODES; ++j) deg += A[tid * N_NODES + j];
  dinv[tid] = (deg > 0.0f) ? (1.0f / sqrtf(fmaxf(deg, 1e-12f))) : 0.0f;
  __syncthreads();
  for (int j = 0; j < N_NODES; ++j) {
    const float v =
        ((tid == j) ? 1.0f : 0.0f) - dinv[tid] * A[tid * N_NODES + j] * dinv[j];
    L[(size_t)g * NN + tid * N_NODES + j] = v;
  }
}

// ---------------------------------------------------------------------------
// 3) filter bank: D_f = Cn[f]*sqrt(2)*a * inv((L-bI)^4 + aI)
//    1 block per (graph,filter); everything LDS-resident (256 KB of 320 KB).
// ---------------------------------------------------------------------------
__global__ void k_filter(const float* __restrict__ L, const float* __restrict__ C,
                         float* __restrict__ Dbuf) {
  extern __shared__ float smem[];
  float* M1 = smem;              // 16384
  float* P2 = smem + NN;         // 16384
  float* AUG = smem + 2 * NN;    // 128 x 256 augmented [P4+aI | I]
  float* sc = smem + 4 * NN;     // [0]=pivot inv, [1]=output coefficient
  const int g = blockIdx.x / NFILT, f = blockIdx.x % NFILT;
  const int tid = threadIdx.x;
  const float bval = 0.1f * (float)f;

  if (tid == 0) {
    float l2 = 0.0f;
    for (int k = 0; k < NFILT; ++k) l2 += C[k] * C[k];
    l2 = fmaxf(sqrtf(l2), EPSV);
    sc[1] = (C[f] / l2) * SQRT2F * A_CONST;  // Cn[f]*sqrt(2)*a
  }
  for (int i = tid; i < NN; i += 256) {
    float v = L[(size_t)g * NN + i];
    const int row = i >> 7, col = i & 127;
    if (row == col) v -= bval;
    M1[i] = v;
  }
  __syncthreads();

  gemm128_lds(M1, M1, P2, 128, 0.0f, tid);  // P2 = M1 @ M1
  __syncthreads();

  for (int i = tid; i < NN; i += 256) {      // right half of AUG = I
    const int row = i >> 7, col = i & 127;
    AUG[row * 256 + 128 + col] = (row == col) ? 1.0f : 0.0f;
  }
  gemm128_lds(P2, P2, AUG, 256, A_CONST, tid);  // left half = P4 + aI
  __syncthreads();

  // Gauss-Jordan, no pivoting (matrix is SPD: sym^4 + aI, a > 0)
  const int irow = tid >> 1, half = tid & 1;
  for (int k = 0; k < N_NODES; ++k) {
    if (tid == 0) sc[0] = 1.0f / AUG[k * 256 + k];
    __syncthreads();
    AUG[k * 256 + tid] *= sc[0];  // scale pivot row (one col / thread)
    __syncthreads();
    const float fv = AUG[irow * 256 + k];  // col k untouched below (j>k only)
    if (irow != k) {
      const float* prow = AUG + k * 256;
      float* rrow = AUG + irow * 256;
      const int j0 = half ? 128 : (k + 1);
      const int j1 = half ? 256 : 128;
      for (int j = j0; j < j1; ++j) rrow[j] -= fv * prow[j];
    }
    __syncthreads();
  }

  const float coef = sc[1];
  for (int i = tid; i < NN; i += 256) {
    const int row = i >> 7, col = i & 127;
    Dbuf[((size_t)g * NFILT + f) * NN + i] = coef * AUG[row * 256 + 128 + col];
  }
}

// ---------------------------------------------------------------------------
// 4) Ahat = I - sum_f Dbuf[g,f]   (deterministic reduction, no atomics)
// ---------------------------------------------------------------------------
__global__ void k_ahat(const float* __restrict__ Dbuf, float* __restrict__ Ahat) {
  const size_t total = (size_t)B_GRAPHS * NN;
  for (size_t idx = blockIdx.x * 256ull + threadIdx.x; idx < total;
       idx += (size_t)gridDim.x * 256ull) {
    const size_t g = idx / NN;
    const int rem = (int)(idx % NN);
    const int row = rem >> 7, col = rem & 127;
    float s = 0.0f;
    const float* base = Dbuf + (size_t)g * NFILT * NN + rem;
    for (int f = 0; f < NFILT; ++f) s += base[(size_t)f * NN];
    Ahat[idx] = ((row == col) ? 1.0f : 0.0f) - s;
  }
}

// ---------------------------------------------------------------------------
// 5) y = Ahat @ feats  (128x128 x 128x1150), WMMA, Ahat in LDS
//    grid = B * 5 blocks; block covers 256 output columns (16 tiles)
// ---------------------------------------------------------------------------
__global__ void k_y(const float* __restrict__ Ahat, const float* __restrict__ feats,
                    float* __restrict__ y) {
  extern __shared__ float As[];
  const int g = blockIdx.x / 5;
  const int cg = blockIdx.x % 5;
  const int tid = threadIdx.x;
  for (int i = tid; i < NN; i += 256) As[i] = Ahat[(size_t)g * NN + i];
  __syncthreads();
  const int lane = tid & 31, wave = tid >> 5, l16 = lane & 15, hi = lane >> 4;
  const float* F = feats + (size_t)g * N_NODES * D_DIM;
  float* Y = y + (size_t)g * N_NODES * D_DIM;
  for (int ct = 0; ct < 2; ++ct) {
    const int col0 = cg * 256 + (wave * 2 + ct) * 16;
    for (int mt = 0; mt < 8; ++mt) {
      v8f acc = {};
      for (int kk = 0; kk < 32; ++kk) {
        const int kb = kk * 4 + 2 * hi;
        v2f a, b;
        const int r = mt * 16 + l16;
        a.x = As[r * 128 + kb];
        a.y = As[r * 128 + kb + 1];
        const int cc = col0 + l16;
        b.x = (cc < D_DIM) ? F[(size_t)kb * D_DIM + cc] : 0.0f;
        b.y = (cc < D_DIM) ? F[(size_t)(kb + 1) * D_DIM + cc] : 0.0f;
        acc = wmma4(a, b, acc);
      }
      for (int v = 0; v < 8; ++v) {
        const int row = mt * 16 + v + 8 * hi;
        const int cc = col0 + l16;
        if (cc < D_DIM) Y[(size_t)row * D_DIM + cc] = acc[v];
      }
    }
  }
}

// ---------------------------------------------------------------------------
// 6) emb[g,d] = mean_n y[g,n,d]
// ---------------------------------------------------------------------------
__global__ void k_emb(const float* __restrict__ y, float* __restrict__ emb) {
  const size_t total = (size_t)B_GRAPHS * D_DIM;
  for (size_t idx = blockIdx.x * 256ull + threadIdx.x; idx < total;
       idx += (size_t)gridDim.x * 256ull) {
    const size_t g = idx / D_DIM;
    const int d = (int)(idx % D_DIM);
    float s = 0.0f;
    const float* Yg = y + (size_t)g * N_NODES * D_DIM + d;
    for (int n = 0; n < N_NODES; ++n) s += Yg[(size_t)n * D_DIM];
    emb[idx] = s / (float)N_NODES;
  }
}

// ---------------------------------------------------------------------------
// 7) Gram: G = y @ y^T per graph (K=1150, zero-padded ragged edge), WMMA
// ---------------------------------------------------------------------------
__global__ void k_gram(const float* __restrict__ y, float* __restrict__ G) {
  const int g = blockIdx.x;
  const int tid = threadIdx.x, lane = tid & 31, wave = tid >> 5;
  const int l16 = lane & 15, hi = lane >> 4;
  const float* Yg = y + (size_t)g * N_NODES * D_DIM;
  for (int t8 = 0; t8 < 8; ++t8) {
    const int t = wave * 8 + t8;
    const int mt = (t >> 3) * 16, nt = (t & 7) * 16;
    v8f acc = {};
    for (int kk = 0; kk < 288; ++kk) {  // ceil(1150/4)
      const int kb = kk * 4 + 2 * hi;
      v2f a, b;
      const int r = mt + l16, cN = nt + l16;
      a.x = (kb < D_DIM) ? Yg[(size_t)r * D_DIM + kb] : 0.0f;
      a.y = (kb + 1 < D_DIM) ? Yg[(size_t)r * D_DIM + kb + 1] : 0.0f;
      b.x = (kb < D_DIM) ? Yg[(size_t)cN * D_DIM + kb] : 0.0f;
      b.y = (kb + 1 < D_DIM) ? Yg[(size_t)cN * D_DIM + kb + 1] : 0.0f;
      acc = wmma4(a, b, acc);
    }
    for (int v = 0; v < 8; ++v) {
      const int row = mt + v + 8 * hi, col = nt + l16;
      G[(size_t)g * NN + row * 128 + col] = acc[v];
    }
  }
}

// ---------------------------------------------------------------------------
// 8) sp[g] = -sum_ij |G_ij| / (c_i c_j) / N^2,  c_i = max(sqrt(G_ii), EPS)
// ---------------------------------------------------------------------------
__global__ void k_sp(const float* __restrict__ G, float* __restrict__ sp) {
  __shared__ float cn[N_NODES];
  __shared__ float red[256];
  const int g = blockIdx.x, tid = threadIdx.x;
  if (tid < N_NODES) {
    const float gii = G[(size_t)g * NN + tid * N_NODES + tid];
    cn[tid] = fmaxf(sqrtf(fmaxf(gii, 0.0f)), EPSV);
  }
  __syncthreads();
  float local = 0.0f;
  for (int i = tid; i < NN; i += 256) {
    const int r = i >> 7, c = i & 127;
    local += fabsf(G[(size_t)g * NN + i]) / (cn[r] * cn[c]);
  }
  red[tid] = local;
  __syncthreads();
  for (int s = 128; s > 0; s >>= 1) {
    if (tid < s) red[tid] += red[tid + s];
    __syncthreads();
  }
  if (tid == 0) sp[g] = -red[0] / (float)NN;
}

// ---------------------------------------------------------------------------
// 9) final scalar: hinge terms + sparsity_fwd + sparsity_c
// ---------------------------------------------------------------------------
__global__ void k_final(const float* __restrict__ C, const float* __restrict__ sp,
                        const float* __restrict__ emb, const int* __restrict__ nclsP,
                        float* __restrict__ out) {
  __shared__ float red[256];
  __shared__ float accs[2];  // [0]=hl1, [1]=hl2
  const int tid = threadIdx.x;
  int nc = nclsP[0];
  if (nc < 2) nc = 2;
  if (tid == 0) { accs[0] = 0.0f; accs[1] = 0.0f; }
  __syncthreads();
  for (int c = 0; c < nc; ++c) {
    const int P = (B_GRAPHS + nc - 1 - c) / nc;
    const int Q = B_GRAPHS - P;
    // positive group pairwise mean distance
    float local = 0.0f;
    for (int p = tid; p < P * P; p += 256) {
      const int i = c + nc * (p / P);
      const int j = c + nc * (p % P);
      if (i != j) {
        const float* ei = emb + (size_t)i * D_DIM;
        const float* ej = emb + (size_t)j * D_DIM;
        float d2 = 0.0f;
        for (int d = 0; d < D_DIM; ++d) {
          const float df = ei[d] - ej[d];
          d2 += df * df;
        }
        if (d2 > 0.0f) local += sqrtf(d2);
      }
    }
    red[tid] = local;
    __syncthreads();
    for (int s = 128; s > 0; s >>= 1) {
      if (tid < s) red[tid] += red[tid + s];
      __syncthreads();
    }
    const float posMean = red[0] / (float)(P * P);
    __syncthreads();
    // negative group
    local = 0.0f;
    const int g1 = nc - 1;
    for (int p = tid; p < Q * Q; p += 256) {
      const int m1 = p / Q, m2 = p % Q;
      int r1 = m1 % g1; r1 += (r1 >= c) ? 1 : 0;
      int r2 = m2 % g1; r2 += (r2 >= c) ? 1 : 0;
      const int i = (m1 / g1) * nc + r1;
      const int j = (m2 / g1) * nc + r2;
      if (i != j) {
        const float* ei = emb + (size_t)i * D_DIM;
        const float* ej = emb + (size_t)j * D_DIM;
        float d2 = 0.0f;
        for (int d = 0; d < D_DIM; ++d) {
          const float df = ei[d] - ej[d];
          d2 += df * df;
        }
        if (d2 > 0.0f) local += sqrtf(d2);
      }
    }
    red[tid] = local;
    __syncthreads();
    for (int s = 128; s > 0; s >>= 1) {
      if (tid < s) red[tid] += red[tid + s];
      __syncthreads();
    }
    if (tid == 0) {
      const float negMean = red[0] / (float)(Q * Q);
      const float beta = 1.0f / (float)nc + 1e-13f;
      accs[0] += -negMean / beta;
      accs[1] += posMean;
    }
    __syncthreads();
  }
  if (tid == 0) {
    // sparsity_fwd = sum_g sp[g] * B^{-(B-g)}
    float spf = 0.0f;
    const float lb = logf((float)B_GRAPHS);
    for (int g = 0; g < B_GRAPHS; ++g)
      spf += sp[g] * expf(-(float)(B_GRAPHS - g) * lb);
    // sparsity_c
    float l1 = 0.0f, l2 = 0.0f;
    for (int f = 0; f < NFILT; ++f) {
      l1 += fabsf(C[f]);
      l2 += C[f] * C[f];
    }
    l2 = fmaxf(sqrtf(l2), EPSV);
    const float dims = sqrtf((float)NFILT);
    const float spc = (dims - l1 / l2) / (dims - 1.0f);
    out[0] = spc + accs[1] + accs[0] + spf;
  }
}

// ---------------------------------------------------------------------------
extern "C" void kernel_launch(void* const* d_in, const int* in_sizes, int n_in,
                              void* d_out, int out_size, void* d_ws, size_t ws_size,
                              hipStream_t stream) {
  (void)in_sizes; (void)n_in; (void)out_size; (void)ws_size;
  const float* x = (const float*)d_in[0];
  const float* rs = (const float*)d_in[1];
  const float* C = (const float*)d_in[2];
  const int* esrc = (const int*)d_in[3];
  const int* edst = (const int*)d_in[4];
  const int* ncls = (const int*)d_in[5];
  float* out = (float*)d_out;
  float* ws = (float*)d_ws;

  // workspace layout (floats)
  float* feats = ws;                                          // B*N*D
  float* L = feats + (size_t)B_GRAPHS * N_NODES * D_DIM;      // B*N*N
  float* Dbuf = L + (size_t)B_GRAPHS * NN;                    // B*21*N*N
  float* Ahat = Dbuf + (size_t)B_GRAPHS * NFILT * NN;         // B*N*N
  float* Y = Ahat + (size_t)B_GRAPHS * NN;                    // B*N*D
  float* G = Y + (size_t)B_GRAPHS * N_NODES * D_DIM;          // B*N*N
  float* emb = G + (size_t)B_GRAPHS * NN;                     // B*D
  float* sp = emb + (size_t)B_GRAPHS * D_DIM;                 // B

  const int totF = B_GRAPHS * N_NODES * D_DIM;
  k_feats<<<(totF + 255) / 256, 256, 0, stream>>>(x, rs, feats);
  k_lap<<<B_GRAPHS, 128, (NN + 128) * sizeof(float), stream>>>(esrc, edst, L);
  k_filter<<<B_GRAPHS * NFILT, 256, (4 * NN + 2) * sizeof(float), stream>>>(L, C, Dbuf);
  const int totA = B_GRAPHS * NN;
  k_ahat<<<(totA + 255) / 256, 256, 0, stream>>>(Dbuf, Ahat);
  k_y<<<B_GRAPHS * 5, 256, NN * sizeof(float), stream>>>(Ahat, feats, Y);
  const int totE = B_GRAPHS * D_DIM;
  k_emb<<<(totE + 255) / 256, 256, 0, stream>>>(Y, emb);
  k_gram<<<B_GRAPHS, 256, 0, stream>>>(Y, G);
  k_sp<<<B_GRAPHS, 256, 0, stream>>>(G, sp);
  k_final<<<1, 256, 0, stream>>>(C, sp, emb, ncls, out);
}